// SRR_53850299957258
// MI455X (gfx1250) — compile-verified
//
#include <hip/hip_runtime.h>
#include <math.h>

// ---------------------------------------------------------------------------
// Slice-to-volume reconstruction (SRR) with 2-iter CG on MI455X (gfx1250).
// N_SLICES=64, H=W=64, VD=VH=VW=64, K=3 (27 PSF taps), RES=1.5.
// All arrays are L2-resident (1 MB each); f32 throughout.
// PSF contraction runs on the matrix pipe via V_WMMA_F32_16X16X4_F32
// (16 pixels / wave32, 7 chained k-chunks, fully unrolled).
// Per-slice rotated PSF offsets {R*off_k, psf_k} are staged in LDS once per
// block, turning per-sample offset math into one ds_load_b128 + 3 adds.
// ---------------------------------------------------------------------------

#define NVOX (64 * 64 * 64)   // 262144, also S*H*W

typedef float v2f __attribute__((ext_vector_type(2)));
typedef float v8f __attribute__((ext_vector_type(8)));

// ---- helpers ---------------------------------------------------------------

__device__ __forceinline__ void k_offsets(int k, float& ox, float& oy, float& oz) {
  int kz  = k / 9;
  int rem = k - kz * 9;
  int ky  = rem / 3;
  int kx  = rem - ky * 3;
  ox = (float)(kx - 1);
  oy = (float)(ky - 1);
  oz = (float)(kz - 1);
}

__device__ __forceinline__ int clamp63(int v) {
  return v < 0 ? 0 : (v > 63 ? 63 : v);
}

// trilinear gather with reference-matching OOB behavior (weight=0, clamped idx)
__device__ __forceinline__ float trisample(const float* __restrict__ vol,
                                           float px, float py, float pz) {
  float xf = floorf(px), yf = floorf(py), zf = floorf(pz);
  int   x0 = (int)xf,    y0 = (int)yf,    z0 = (int)zf;
  float fx = px - xf, fy = py - yf, fz = pz - zf;
  float gx[2] = {1.0f - fx, fx};
  float gy[2] = {1.0f - fy, fy};
  float gz[2] = {1.0f - fz, fz};
  float acc = 0.0f;
#pragma unroll
  for (int dz = 0; dz < 2; ++dz) {
    int  zi = z0 + dz;
    bool vz = (zi >= 0) & (zi < 64);
    int  zc = clamp63(zi);
#pragma unroll
    for (int dy = 0; dy < 2; ++dy) {
      int  yi = y0 + dy;
      bool vy = (yi >= 0) & (yi < 64);
      int  yc = clamp63(yi);
#pragma unroll
      for (int dx = 0; dx < 2; ++dx) {
        int  xi = x0 + dx;
        bool vx = (xi >= 0) & (xi < 64);
        int  xc = clamp63(xi);
        float w = gx[dx] * gy[dy] * gz[dz];
        w = (vx & vy & vz) ? w : 0.0f;
        acc += w * vol[(zc << 12) + (yc << 6) + xc];
      }
    }
  }
  return acc;
}

__device__ __forceinline__ void triscatter(float* __restrict__ vol,
                                           float px, float py, float pz, float val) {
  float xf = floorf(px), yf = floorf(py), zf = floorf(pz);
  int   x0 = (int)xf,    y0 = (int)yf,    z0 = (int)zf;
  float fx = px - xf, fy = py - yf, fz = pz - zf;
  float gx[2] = {1.0f - fx, fx};
  float gy[2] = {1.0f - fy, fy};
  float gz[2] = {1.0f - fz, fz};
#pragma unroll
  for (int dz = 0; dz < 2; ++dz) {
    int  zi = z0 + dz;
    bool vz = (zi >= 0) & (zi < 64);
#pragma unroll
    for (int dy = 0; dy < 2; ++dy) {
      int  yi = y0 + dy;
      bool vy = (yi >= 0) & (yi < 64);
#pragma unroll
      for (int dx = 0; dx < 2; ++dx) {
        int  xi = x0 + dx;
        bool vx = (xi >= 0) & (xi < 64);
        if (vx & vy & vz) {
          float w = gx[dx] * gy[dy] * gz[dz];
          atomicAdd(&vol[(zi << 12) + (yi << 6) + xi], w * val);
        }
      }
    }
  }
}

// LDS tree reduction + one f32 atomic per block
__device__ __forceinline__ void block_reduce_atomic(float v, float* target) {
  __shared__ float sm[256];
  sm[threadIdx.x] = v;
  __syncthreads();
#pragma unroll
  for (int off = 128; off > 0; off >>= 1) {
    if ((int)threadIdx.x < off) sm[threadIdx.x] += sm[threadIdx.x + off];
    __syncthreads();
  }
  if (threadIdx.x == 0) atomicAdd(target, sm[0]);
}

// Stage per-slice rotated PSF offsets into LDS: koff[k] = {R*off_k, psf[k]}.
// npad entries beyond 26 are zero-filled (psf weight 0 kills their effect).
__device__ __forceinline__ void stage_koff(float4* koff, int npad,
                                           const float* __restrict__ psf,
                                           float r00, float r01, float r02,
                                           float r10, float r11, float r12,
                                           float r20, float r21, float r22) {
  if ((int)threadIdx.x < npad) {
    int k = threadIdx.x;
    float4 e = {0.0f, 0.0f, 0.0f, 0.0f};
    if (k < 27) {
      float ox, oy, oz;
      k_offsets(k, ox, oy, oz);
      e.x = r00 * ox + r01 * oy + r02 * oz;
      e.y = r10 * ox + r11 * oy + r12 * oz;
      e.z = r20 * ox + r21 * oy + r22 * oz;
      e.w = psf[k];
    }
    koff[k] = e;
  }
  __syncthreads();
}

// ---- A operator: gather + WMMA PSF contraction -----------------------------
// grid = 64 slices * 32 blocks, block = 256 (8 waves). Each wave -> 16 pixels.
// Lane l handles pixel (l&15); lanes 0-15 supply k%4 in {0,1}, lanes 16-31
// supply k%4 in {2,3}. A(16x4)=psf replicated rows, B(4x16)=samples,
// D accumulated across 7 chunks; all rows of D identical -> d[0] is result.
__global__ __launch_bounds__(256) void a_gather_wmma(
    const float* __restrict__ vol, const float* __restrict__ T,
    const float* __restrict__ psf, float* __restrict__ out) {
  __shared__ float4 koff[28];

  const int lane = threadIdx.x & 31;
  const int wave = threadIdx.x >> 5;
  const int s    = blockIdx.x >> 5;                      // 32 blocks per slice
  const int pix  = ((blockIdx.x & 31) * 8 + wave) * 16 + (lane & 15);
  const int h = pix >> 6, w = pix & 63;

  const float* Ts = T + s * 12;                          // s-uniform -> s_loads
  float r00 = Ts[0], r01 = Ts[1], r02 = Ts[2],  tx = Ts[3];
  float r10 = Ts[4], r11 = Ts[5], r12 = Ts[6],  ty = Ts[7];
  float r20 = Ts[8], r21 = Ts[9], r22 = Ts[10], tz = Ts[11];

  stage_koff(koff, 28, psf, r00, r01, r02, r10, r11, r12, r20, r21, r22);

  float xs = ((float)w - 31.5f) * 1.5f;
  float ys = ((float)h - 31.5f) * 1.5f;
  float qx = r00 * xs + r01 * ys + tx + 31.5f;
  float qy = r10 * xs + r11 * ys + ty + 31.5f;
  float qz = r20 * xs + r21 * ys + tz + 31.5f;

  const int kbase = (lane >> 4) << 1;                    // 0 (lo half) or 2
  v8f d = {0.f, 0.f, 0.f, 0.f, 0.f, 0.f, 0.f, 0.f};

#pragma unroll
  for (int c = 0; c < 7; ++c) {
    float4 o0 = koff[4 * c + kbase];
    float4 o1 = koff[4 * c + kbase + 1];                 // index 27 = padded 0
    float b0 = trisample(vol, qx + o0.x, qy + o0.y, qz + o0.z);
    float b1 = trisample(vol, qx + o1.x, qy + o1.y, qz + o1.z);
    v2f av = {o0.w, o1.w};
    v2f bv = {b0, b1};
    d = __builtin_amdgcn_wmma_f32_16x16x4_f32(false, av, false, bv,
                                              (short)0, d, false, false);
  }
  if (lane < 16) out[(s << 12) + pix] = d[0];
}

// ---- At operator: scatter with atomics -------------------------------------
__global__ __launch_bounds__(256) void at_scatter(
    const float* __restrict__ src, const float* __restrict__ T,
    const float* __restrict__ psf, float* __restrict__ vol) {
  __shared__ float4 koff[27];

  int i = blockIdx.x * 256 + threadIdx.x;                // 0..NVOX-1
  int s = i >> 12;                                       // 16 blocks per slice
  int p = i & 4095;
  int h = p >> 6, w = p & 63;

  const float* Ts = T + s * 12;
  float r00 = Ts[0], r01 = Ts[1], r02 = Ts[2],  tx = Ts[3];
  float r10 = Ts[4], r11 = Ts[5], r12 = Ts[6],  ty = Ts[7];
  float r20 = Ts[8], r21 = Ts[9], r22 = Ts[10], tz = Ts[11];

  stage_koff(koff, 27, psf, r00, r01, r02, r10, r11, r12, r20, r21, r22);

  float xs = ((float)w - 31.5f) * 1.5f;
  float ys = ((float)h - 31.5f) * 1.5f;
  float qx = r00 * xs + r01 * ys + tx + 31.5f;
  float qy = r10 * xs + r11 * ys + ty + 31.5f;
  float qz = r20 * xs + r21 * ys + tz + 31.5f;

  float v = src[i];
#pragma unroll 1
  for (int k = 0; k < 27; ++k) {
    float4 o = koff[k];
    triscatter(vol, qx + o.x, qy + o.y, qz + o.z, o.w * v);
  }
}

// ---- CG plumbing -----------------------------------------------------------

__global__ __launch_bounds__(256) void init_k(const float* __restrict__ vol0,
                                              float* __restrict__ x,
                                              float* __restrict__ b,
                                              float* __restrict__ Ap,
                                              float* __restrict__ sc) {
  int i = blockIdx.x * 256 + threadIdx.x;
  x[i]  = vol0[i];
  b[i]  = 0.0f;
  Ap[i] = 0.0f;
  if (i < 8) sc[i] = 0.0f;
}

// r = b - Ap; p = r; sc[0] += r.r; Ap zeroed for next use
__global__ __launch_bounds__(256) void cg_init(const float* __restrict__ b,
                                               float* __restrict__ Ap,
                                               float* __restrict__ r,
                                               float* __restrict__ p,
                                               float* __restrict__ sc) {
  int i = blockIdx.x * 256 + threadIdx.x;
  float rv = b[i] - Ap[i];
  Ap[i] = 0.0f;
  r[i] = rv;
  p[i] = rv;
  block_reduce_atomic(rv * rv, &sc[0]);
}

__global__ __launch_bounds__(256) void dot_k(const float* __restrict__ a,
                                             const float* __restrict__ b,
                                             float* __restrict__ target) {
  int i = blockIdx.x * 256 + threadIdx.x;
  block_reduce_atomic(a[i] * b[i], target);
}

// alpha = sc[0]/sc[1]; x += a*p; r -= a*Ap; sc[2] += r.r; Ap zeroed
__global__ __launch_bounds__(256) void cg_update0(float* __restrict__ x,
                                                  float* __restrict__ r,
                                                  const float* __restrict__ p,
                                                  float* __restrict__ Ap,
                                                  float* __restrict__ sc) {
  int i = blockIdx.x * 256 + threadIdx.x;
  float alpha = sc[0] / sc[1];
  x[i] = x[i] + alpha * p[i];
  float rn = r[i] - alpha * Ap[i];
  Ap[i] = 0.0f;
  r[i] = rn;
  block_reduce_atomic(rn * rn, &sc[2]);
}

// beta = sc[2]/sc[0]; p = r + beta*p
__global__ __launch_bounds__(256) void cg_updatep(float* __restrict__ p,
                                                  const float* __restrict__ r,
                                                  const float* __restrict__ sc) {
  int i = blockIdx.x * 256 + threadIdx.x;
  float beta = sc[2] / sc[0];
  p[i] = r[i] + beta * p[i];
}

// alpha = sc[2]/sc[3]; out = relu(x + alpha*p)
__global__ __launch_bounds__(256) void finalize_k(const float* __restrict__ x,
                                                  const float* __restrict__ p,
                                                  const float* __restrict__ sc,
                                                  float* __restrict__ out) {
  int i = blockIdx.x * 256 + threadIdx.x;
  float alpha = sc[2] / sc[3];
  float v = x[i] + alpha * p[i];
  out[i] = v > 0.0f ? v : 0.0f;
}

// ---- driver ----------------------------------------------------------------

extern "C" void kernel_launch(void* const* d_in, const int* in_sizes, int n_in,
                              void* d_out, int out_size, void* d_ws, size_t ws_size,
                              hipStream_t stream) {
  const float* T    = (const float*)d_in[0];   // (64,3,4)
  const float* slc  = (const float*)d_in[1];   // (64,64,64)
  const float* vol0 = (const float*)d_in[2];   // (64,64,64)
  const float* psf  = (const float*)d_in[3];   // (27,)

  float* x = (float*)d_out;                    // x lives in d_out

  float* ws  = (float*)d_ws;
  float* b   = ws;                             // 1 MB each
  float* tmp = ws + 1 * NVOX;
  float* Ap  = ws + 2 * NVOX;
  float* r   = ws + 3 * NVOX;
  float* p   = ws + 4 * NVOX;
  float* sc  = ws + 5 * NVOX;                  // scalars: rr0, pAp0, rr1, pAp1

  dim3 blk(256);
  dim3 grdE(NVOX / 256);   // 1024 blocks: elementwise / scatter
  dim3 grdG(64 * 32);      // 2048 blocks: WMMA gather (128 px / block)

  // x = vol0; b = 0; Ap = 0; scalars = 0
  init_k<<<grdE, blk, 0, stream>>>(vol0, x, b, Ap, sc);
  // b = At(slices)
  at_scatter<<<grdE, blk, 0, stream>>>(slc, T, psf, b);
  // Ap = AtA(x)
  a_gather_wmma<<<grdG, blk, 0, stream>>>(x, T, psf, tmp);
  at_scatter<<<grdE, blk, 0, stream>>>(tmp, T, psf, Ap);
  // r = b - Ap; p = r; rr0 = <r,r>; Ap cleared
  cg_init<<<grdE, blk, 0, stream>>>(b, Ap, r, p, sc);
  // iteration 0: Ap = AtA(p); alpha = rr0/<p,Ap>; x,r update; rr1; p update
  a_gather_wmma<<<grdG, blk, 0, stream>>>(p, T, psf, tmp);
  at_scatter<<<grdE, blk, 0, stream>>>(tmp, T, psf, Ap);
  dot_k<<<grdE, blk, 0, stream>>>(p, Ap, sc + 1);
  cg_update0<<<grdE, blk, 0, stream>>>(x, r, p, Ap, sc);
  cg_updatep<<<grdE, blk, 0, stream>>>(p, r, sc);
  // iteration 1: Ap = AtA(p); alpha = rr1/<p,Ap>; out = relu(x + alpha p)
  a_gather_wmma<<<grdG, blk, 0, stream>>>(p, T, psf, tmp);
  at_scatter<<<grdE, blk, 0, stream>>>(tmp, T, psf, Ap);
  dot_k<<<grdE, blk, 0, stream>>>(p, Ap, sc + 3);
  finalize_k<<<grdE, blk, 0, stream>>>(x, p, sc, x);
}